// ExonIntronModel_3272765080023
// MI455X (gfx1250) — compile-verified
//
#include <hip/hip_runtime.h>

typedef float v2f __attribute__((ext_vector_type(2)));
typedef float v8f __attribute__((ext_vector_type(8)));

#define B_   8
#define T_   8192
#define DNA_ 4
#define HID_ 256
#define C_   5
#define K_   16
#define NEGV (-1e30f)

__device__ __forceinline__ float fast_tanh(float x) {
    float e = __expf(2.0f * x);
    return (e - 1.0f) / (e + 1.0f);
}

// jax.nn.gelu default (approximate=True, tanh form)
__device__ __forceinline__ float gelu_tanh(float x) {
    const float k0 = 0.7978845608028654f;  // sqrt(2/pi)
    const float k1 = 0.044715f;
    float t = fast_tanh(k0 * (x + k1 * x * x * x));
    return 0.5f * x * (1.0f + t);
}

// ---------------------------------------------------------------------------
// Kernel 1: emit = gelu(seq @ W_enc + b_enc) @ W_proj + b_proj
// One wave per 16-row tile. GEMM1 uses V_WMMA_F32_16X16X4_F32 (K=4 == DNA_DIM,
// exact f32). GEMM2 (HID=256 -> C=5) is fused: WMMA D-layout gives each lane
// one hidden column n, so partials are FMAs + a 16-lane xor-butterfly over n.
// ---------------------------------------------------------------------------
__global__ __launch_bounds__(256) void encoder_emit_kernel(
    const float* __restrict__ seq,     // (B*T, 4) row-major
    const float* __restrict__ W_enc,   // (4, 256)
    const float* __restrict__ b_enc,   // (256)
    const float* __restrict__ W_proj,  // (256, 5)
    const float* __restrict__ b_proj,  // (5)
    float* __restrict__ emit)          // (B*T, 5)
{
    const int lane = threadIdx.x & 31;
    const int wave = threadIdx.x >> 5;
    const int tile = blockIdx.x * 8 + wave;   // 16-row tile id (4096 total)
    const long r0  = (long)tile * 16;

    const int half = lane >> 4;   // 0: lanes 0-15, 1: lanes 16-31
    const int l16  = lane & 15;

    // A tile (16x4 f32): lane l<16 -> M=l, VGPRs hold K={0,1}; lanes 16-31 K={2,3}
    v2f a;
    {
        const float* p = seq + (r0 + l16) * DNA_ + 2 * half;
        a.x = p[0];
        a.y = p[1];
    }

    float acc[8][C_];
#pragma unroll
    for (int v = 0; v < 8; ++v)
#pragma unroll
        for (int c = 0; c < C_; ++c) acc[v][c] = 0.0f;

#pragma unroll
    for (int nt = 0; nt < HID_ / 16; ++nt) {
        const int n = nt * 16 + l16;

        // B tile (4x16 f32): lane -> N=l16, VGPRs hold K={0,1} (low half) / {2,3}
        v2f bb;
        bb.x = W_enc[(2 * half + 0) * HID_ + n];
        bb.y = W_enc[(2 * half + 1) * HID_ + n];

        // C init with bias: every element of this lane has column n
        v8f cc;
        {
            float be = b_enc[n];
#pragma unroll
            for (int v = 0; v < 8; ++v) cc[v] = be;
        }

        // D = A*B + C  (exact f32 WMMA, K=4)
        cc = __builtin_amdgcn_wmma_f32_16x16x4_f32(
                 false, a, false, bb, (short)0, cc, false, false);

        // D layout: cc[v] = hidden[m = v + 8*half][n].  Fused projection partials.
        float wp[C_];
#pragma unroll
        for (int c = 0; c < C_; ++c) wp[c] = W_proj[n * C_ + c];
#pragma unroll
        for (int v = 0; v < 8; ++v) {
            float h = gelu_tanh(cc[v]);
#pragma unroll
            for (int c = 0; c < C_; ++c)
                acc[v][c] = fmaf(h, wp[c], acc[v][c]);
        }
    }

    // Sum over n: butterfly within each 16-lane half (masks 1,2,4,8 stay in-half)
#pragma unroll
    for (int v = 0; v < 8; ++v)
#pragma unroll
        for (int c = 0; c < C_; ++c) {
            float s = acc[v][c];
            s += __shfl_xor(s, 1, 32);
            s += __shfl_xor(s, 2, 32);
            s += __shfl_xor(s, 4, 32);
            s += __shfl_xor(s, 8, 32);
            acc[v][c] = s;
        }

    // lanes 0 / 16 hold full sums for rows r0+{0..7} / r0+{8..15}
    if (l16 == 0) {
#pragma unroll
        for (int v = 0; v < 8; ++v) {
            const long row = r0 + half * 8 + v;
#pragma unroll
            for (int c = 0; c < C_; ++c)
                emit[row * C_ + c] = acc[v][c] + b_proj[c];
        }
    }
}

// ---------------------------------------------------------------------------
// Kernel 2: cumsum + semi-Markov CRF forward scan. One wave per batch.
// Lane j (0..15) = duration slot: holds H[j][c] and ring value cum[t-j][c].
// Per step: shfl_xor butterfly logsumexp over durations, tiny 5x5 transition
// logsumexp in-lane, shfl_up rotations for the H / cum-ring shifts.
// Lanes 16-31 execute uniformly (shuffle safety) but their results are unused.
// ---------------------------------------------------------------------------
__global__ __launch_bounds__(32) void crf_scan_kernel(
    const float* __restrict__ emit,     // (B, T, 5)
    const int* __restrict__ lengths,    // (B)
    const float* __restrict__ trans,    // (5, 5)
    const float* __restrict__ start,    // (5)
    const float* __restrict__ dur,      // (5, 16)
    float* __restrict__ out)            // [0..B) partition, then cum (B,T+1,5)
{
    const int b = blockIdx.x;
    const int j = threadIdx.x;          // duration-1 slot for lanes 0..15
    const float* em = emit + (long)b * T_ * C_;
    float* cum_out  = out + B_ + (long)b * (T_ + 1) * C_;
    const int len   = lengths[b];

    float tr[C_][C_];
#pragma unroll
    for (int cp = 0; cp < C_; ++cp)
#pragma unroll
        for (int c = 0; c < C_; ++c) tr[cp][c] = trans[cp * C_ + c];

    float du[C_];
#pragma unroll
    for (int c = 0; c < C_; ++c) du[c] = dur[c * K_ + (j & 15)];

    float H[C_], pc[C_], cumc[C_];
#pragma unroll
    for (int c = 0; c < C_; ++c) {
        H[c]    = (j == 0) ? start[c] : NEGV;
        pc[c]   = 0.0f;   // cum[t-j] ring (cum_pad zeros)
        cumc[c] = 0.0f;   // running cumsum (redundant per lane)
    }

    if (j < C_) cum_out[j] = 0.0f;      // cum row 0 = 0

    for (int t = 0; t < T_; ++t) {
        float cn[C_];
#pragma unroll
        for (int c = 0; c < C_; ++c) {
            cumc[c] += em[(long)t * C_ + c];
            cn[c] = cumc[c];            // cum[t+1]
        }

        // alpha[c] = logsumexp_j ( H[j][c] + (cum[t+1]-cum[t-j])[c] + dur[c][j] )
        float alpha[C_];
#pragma unroll
        for (int c = 0; c < C_; ++c) {
            float s  = H[c] + (cn[c] - pc[c]) + du[c];
            float mx = s;
            mx = fmaxf(mx, __shfl_xor(mx, 1, 32));
            mx = fmaxf(mx, __shfl_xor(mx, 2, 32));
            mx = fmaxf(mx, __shfl_xor(mx, 4, 32));
            mx = fmaxf(mx, __shfl_xor(mx, 8, 32));
            float ex = __expf(s - mx);
            ex += __shfl_xor(ex, 1, 32);
            ex += __shfl_xor(ex, 2, 32);
            ex += __shfl_xor(ex, 4, 32);
            ex += __shfl_xor(ex, 8, 32);
            alpha[c] = mx + __logf(ex);
        }

        // m[c] = logsumexp_{c'} alpha[c'] + trans[c'][c]
        float m[C_];
#pragma unroll
        for (int c = 0; c < C_; ++c) {
            float mx = alpha[0] + tr[0][c];
#pragma unroll
            for (int cp = 1; cp < C_; ++cp)
                mx = fmaxf(mx, alpha[cp] + tr[cp][c]);
            float s = 0.0f;
#pragma unroll
            for (int cp = 0; cp < C_; ++cp)
                s += __expf(alpha[cp] + tr[cp][c] - mx);
            m[c] = mx + __logf(s);
        }

        // shift: H' = [m, H[:-1]], ring' = [cum[t+1], ring[:-1]]
#pragma unroll
        for (int c = 0; c < C_; ++c) {
            float hs = __shfl_up(H[c], 1, 32);
            float ps = __shfl_up(pc[c], 1, 32);
            H[c]  = (j == 0) ? m[c]  : hs;
            pc[c] = (j == 0) ? cn[c] : ps;
        }

        if (j == 0) {
#pragma unroll
            for (int c = 0; c < C_; ++c)
                cum_out[(long)(t + 1) * C_ + c] = cn[c];
            if (t == len - 1) {
                float mx = alpha[0];
#pragma unroll
                for (int c = 1; c < C_; ++c) mx = fmaxf(mx, alpha[c]);
                float s = 0.0f;
#pragma unroll
                for (int c = 0; c < C_; ++c) s += __expf(alpha[c] - mx);
                out[b] = mx + __logf(s);
            }
        }
    }
}

extern "C" void kernel_launch(void* const* d_in, const int* in_sizes, int n_in,
                              void* d_out, int out_size, void* d_ws, size_t ws_size,
                              hipStream_t stream) {
    const float* seq    = (const float*)d_in[0];
    const int*   len    = (const int*)d_in[1];
    const float* W_enc  = (const float*)d_in[2];
    const float* b_enc  = (const float*)d_in[3];
    const float* W_proj = (const float*)d_in[4];
    const float* b_proj = (const float*)d_in[5];
    const float* trans  = (const float*)d_in[6];
    const float* start  = (const float*)d_in[7];
    const float* dur    = (const float*)d_in[8];
    float* out  = (float*)d_out;
    float* emit = (float*)d_ws;   // B*T*C floats = 1.31 MB scratch

    // 65536 rows / 16 per wave = 4096 waves; 8 waves (256 thr) per block.
    encoder_emit_kernel<<<(B_ * T_ / 16) / 8, 256, 0, stream>>>(
        seq, W_enc, b_enc, W_proj, b_proj, emit);

    crf_scan_kernel<<<B_, 32, 0, stream>>>(emit, len, trans, start, dur, out);
}